// SPR_RGCN_OneHot_88648124990481
// MI455X (gfx1250) — compile-verified
//
#include <hip/hip_runtime.h>

typedef __attribute__((ext_vector_type(2))) float v2f;
typedef __attribute__((ext_vector_type(8))) float v8f;

#define NREL 3

// ---------------------------------------------------------------- zero (ints & floats share 0x0)
__global__ void k_zero_i32(int* __restrict__ p, long n) {
  long i = (long)blockIdx.x * blockDim.x + threadIdx.x;
  if (i < n) p[i] = 0;
}

// ---------------------------------------------------------------- one-hot + pre linear + relu
// h0[n, m] = relu(pre_w[shape, m] + pre_w[8 + color, m] + pre_b[m]),  m < 32
__global__ void k_pre_onehot(const int* __restrict__ node_x,
                             const float* __restrict__ pre_w,   // [16,32]
                             const float* __restrict__ pre_b,   // [32]
                             float* __restrict__ h0,            // [N,32]
                             int N) {
  int idx = blockIdx.x * blockDim.x + threadIdx.x;
  int n = idx >> 5, m = idx & 31;
  if (n >= N) return;
  int s = node_x[2 * n + 0];
  int c = node_x[2 * n + 1];
  float v = pre_w[s * 32 + m] + pre_w[(8 + c) * 32 + m] + pre_b[m];
  h0[n * 32 + m] = v > 0.f ? v : 0.f;
}

// ---------------------------------------------------------------- per-relation in-degree counts
__global__ void k_count_edges(const int* __restrict__ dst,
                              const int* __restrict__ etype,
                              int* __restrict__ cnt,            // [3,N]
                              int E, int N) {
  int e = blockIdx.x * blockDim.x + threadIdx.x;
  if (e >= E) return;
  atomicAdd(&cnt[etype[e] * N + dst[e]], 1);
}

__global__ void k_inv_counts(const int* __restrict__ cnt,
                             float* __restrict__ inv, long n) {
  long i = (long)blockIdx.x * blockDim.x + threadIdx.x;
  if (i >= n) return;
  int c = cnt[i];
  inv[i] = 1.0f / (float)(c > 1 ? c : 1);
}

// ---------------------------------------------------------------- fp32 WMMA GEMM: out[N,64] = x[N,Kin] @ w[Kin,64] (+bias)
// One wave = one 16x16 output tile; 4 waves per block cover the 64 output cols.
// A layout (16x4 f32): lane<16 -> row=lane, holds K = k0,k0+1 ; lane>=16 -> row=lane-16, K = k0+2,k0+3.
// B layout (4x16 f32): symmetric, lane%16 = output col, lane/16 selects K pair.
// C/D layout: VGPR j -> row = j + 8*(lane>=16), col = lane%16.
__global__ void __launch_bounds__(128)
k_wmma_gemm_f32(const float* __restrict__ x, int ldx,          // [N, Kin], row stride ldx
                const float* __restrict__ w,                   // [Kin, 64] row-major
                const float* __restrict__ bias,                // [64] or nullptr
                float* __restrict__ out,                       // [N, 64]
                int Kin) {
  const int lane = threadIdx.x & 31;
  const int wave = threadIdx.x >> 5;     // output col tile 0..3
  const int half = lane >> 4;            // 0 or 1
  const int l16  = lane & 15;
  const long row = (long)blockIdx.x * 16 + l16;   // node row for A loads
  const int  col = wave * 16 + l16;               // feature col for B loads

  v8f c = {};
  if (bias) {
    float bv = bias[col];
#pragma unroll
    for (int j = 0; j < 8; ++j) c[j] = bv;
  }

  const float* xa = x + row * ldx + 2 * half;
  const float* wb = w + (2 * half) * 64 + col;

  for (int k0 = 0; k0 < Kin; k0 += 4) {
    v2f a, b;
    a.x = xa[k0];
    a.y = xa[k0 + 1];
    b.x = wb[k0 * 64];
    b.y = wb[(k0 + 1) * 64];
    c = __builtin_amdgcn_wmma_f32_16x16x4_f32(
        /*neg_a=*/false, a, /*neg_b=*/false, b,
        /*c_mod=*/(short)0, c, /*reuse_a=*/false, /*reuse_b=*/false);
  }

  const long nbase = (long)blockIdx.x * 16 + 8 * half;
#pragma unroll
  for (int j = 0; j < 8; ++j) {
    out[(nbase + j) * 64 + col] = c[j];
  }
}

// ---------------------------------------------------------------- edge scatter for one relation
// out[dst, f] += xr[src, f] * inv[r, dst]   (64 lanes per edge)
__global__ void k_edge_scatter(const float* __restrict__ xr,   // [N,64]
                               const int* __restrict__ src,
                               const int* __restrict__ dst,
                               const int* __restrict__ etype,
                               const float* __restrict__ inv,  // [3,N]
                               float* __restrict__ out,        // [N,64]
                               int E, int N, int r) {
  int idx = blockIdx.x * blockDim.x + threadIdx.x;
  int e = idx >> 6, f = idx & 63;
  if (e >= E) return;
  if (etype[e] != r) return;
  int s = src[e], d = dst[e];
  atomicAdd(&out[(long)d * 64 + f], xr[(long)s * 64 + f] * inv[(long)r * N + d]);
}

// ---------------------------------------------------------------- relu in place
__global__ void k_relu(float* __restrict__ p, long n) {
  long i = (long)blockIdx.x * blockDim.x + threadIdx.x;
  if (i >= n) return;
  float v = p[i];
  p[i] = v > 0.f ? v : 0.f;
}

// ---------------------------------------------------------------- global mean pool accumulation
__global__ void k_pool(const float* __restrict__ h,            // [N,64]
                       const int* __restrict__ batch,          // [N]
                       float* __restrict__ gsum,               // [G,64]
                       int* __restrict__ gcnt,                 // [G]
                       int N) {
  int idx = blockIdx.x * blockDim.x + threadIdx.x;
  int n = idx >> 6, f = idx & 63;
  if (n >= N) return;
  int g = batch[n];
  atomicAdd(&gsum[(long)g * 64 + f], h[(long)n * 64 + f]);
  if (f == 0) atomicAdd(&gcnt[g], 1);
}

// ---------------------------------------------------------------- classifier: out[g,c] = mean(h_g) @ cls_w + cls_b
__global__ void k_classify(const float* __restrict__ gsum,     // [G,64]
                           const int* __restrict__ gcnt,       // [G]
                           const float* __restrict__ cls_w,    // [64,10]
                           const float* __restrict__ cls_b,    // [10]
                           float* __restrict__ out,            // [G,10]
                           int G) {
  int idx = blockIdx.x * blockDim.x + threadIdx.x;
  int g = idx / 10, c = idx % 10;
  if (g >= G) return;
  int cn = gcnt[g];
  float invc = 1.0f / (float)(cn > 1 ? cn : 1);
  float acc = cls_b[c];
#pragma unroll
  for (int k = 0; k < 64; ++k)
    acc += gsum[(long)g * 64 + k] * invc * cls_w[k * 10 + c];
  out[idx] = acc;
}

// ================================================================ launch
extern "C" void kernel_launch(void* const* d_in, const int* in_sizes, int n_in,
                              void* d_out, int out_size, void* d_ws, size_t ws_size,
                              hipStream_t stream) {
  const int*   node_x  = (const int*)d_in[0];
  const int*   eidx    = (const int*)d_in[1];
  const int*   etype   = (const int*)d_in[2];
  const int*   batch   = (const int*)d_in[3];
  const float* pre_w   = (const float*)d_in[4];
  const float* pre_b   = (const float*)d_in[5];
  const float* c1_w    = (const float*)d_in[6];   // [3,32,64]
  const float* c1_root = (const float*)d_in[7];   // [32,64]
  const float* c1_b    = (const float*)d_in[8];   // [64]
  const float* c2_w    = (const float*)d_in[9];   // [3,64,64]
  const float* c2_root = (const float*)d_in[10];  // [64,64]
  const float* c2_b    = (const float*)d_in[11];  // [64]
  const float* cls_w   = (const float*)d_in[12];  // [64,10]
  const float* cls_b   = (const float*)d_in[13];  // [10]

  const int N = in_sizes[0] / 2;    // 100000 (multiple of 16)
  const int E = in_sizes[2];        // 1250000
  const int G = out_size / 10;      // 1000
  const int* src = eidx;
  const int* dst = eidx + E;

  // ---- workspace carve-out (~80 MB) ----
  char* ws = (char*)d_ws;
  float* hA   = (float*)ws; ws += (size_t)N * 64 * sizeof(float);
  float* hB   = (float*)ws; ws += (size_t)N * 64 * sizeof(float);
  float* xr   = (float*)ws; ws += (size_t)N * 64 * sizeof(float);
  int*   cnt  = (int*)ws;   ws += (size_t)NREL * N * sizeof(int);
  float* gsum = (float*)ws; ws += (size_t)G * 64 * sizeof(float);
  int*   gcnt = (int*)ws;   ws += (size_t)G * sizeof(int);
  float* inv  = (float*)ws; ws += (size_t)NREL * N * sizeof(float);

  // zero cnt + gsum + gcnt in one pass (contiguous 4-byte words)
  long zn = (long)NREL * N + (long)G * 64 + G;
  k_zero_i32<<<(int)((zn + 255) / 256), 256, 0, stream>>>(cnt, zn);

  // one-hot + pre linear + relu -> hA [N,32]
  k_pre_onehot<<<(N * 32 + 255) / 256, 256, 0, stream>>>(node_x, pre_w, pre_b, hA, N);

  // per-relation in-degree counts & reciprocals (shared by both conv layers)
  k_count_edges<<<(E + 255) / 256, 256, 0, stream>>>(dst, etype, cnt, E, N);
  k_inv_counts<<<(int)(((long)NREL * N + 255) / 256), 256, 0, stream>>>(cnt, inv, (long)NREL * N);

  const int gemm_grid = N / 16;       // 6250 row tiles
  const int scat_grid = (int)(((long)E * 64 + 255) / 256);

  // ---- conv1 : Kin = 32 ----
  k_wmma_gemm_f32<<<gemm_grid, 128, 0, stream>>>(hA, 32, c1_root, c1_b, hB, 32);
  for (int r = 0; r < NREL; ++r) {
    k_wmma_gemm_f32<<<gemm_grid, 128, 0, stream>>>(hA, 32, c1_w + (size_t)r * 32 * 64, nullptr, xr, 32);
    k_edge_scatter<<<scat_grid, 256, 0, stream>>>(xr, src, dst, etype, inv, hB, E, N, r);
  }
  k_relu<<<(int)(((long)N * 64 + 255) / 256), 256, 0, stream>>>(hB, (long)N * 64);

  // ---- conv2 : Kin = 64 ----
  k_wmma_gemm_f32<<<gemm_grid, 128, 0, stream>>>(hB, 64, c2_root, c2_b, hA, 64);
  for (int r = 0; r < NREL; ++r) {
    k_wmma_gemm_f32<<<gemm_grid, 128, 0, stream>>>(hB, 64, c2_w + (size_t)r * 64 * 64, nullptr, xr, 64);
    k_edge_scatter<<<scat_grid, 256, 0, stream>>>(xr, src, dst, etype, inv, hA, E, N, r);
  }
  k_relu<<<(int)(((long)N * 64 + 255) / 256), 256, 0, stream>>>(hA, (long)N * 64);

  // ---- pool + classify ----
  k_pool<<<(int)(((long)N * 64 + 255) / 256), 256, 0, stream>>>(hA, batch, gsum, gcnt, N);
  k_classify<<<(G * 10 + 255) / 256, 256, 0, stream>>>(gsum, gcnt, cls_w, cls_b, (float*)d_out, G);
}